// QuantumFeatureMap_1786706395146
// MI455X (gfx1250) — compile-verified
//
#include <hip/hip_runtime.h>
#include <hip/hip_bf16.h>
#include <math.h>

// ---------------------------------------------------------------------------
// QuantumFeatureMap for MI455X (gfx1250)
//
// Stage 1: h = tanh(relu(X @ W1 + b1) @ W2 + b2) via V_WMMA_F32_16X16X4_F32
// Stage 2: 16-qubit state-vector sim, one workgroup per batch element,
//          full 65536-float state resident in 256KB LDS (CDNA5: 320KB/WGP),
//          93 gate sweeps in LDS, single HBM write of complex64 output.
// The state is provably real (real M, real init, real CNOT mixing), so the
// imaginary part is emitted as 0.
// ---------------------------------------------------------------------------

typedef __attribute__((ext_vector_type(2))) float v2f;
typedef __attribute__((ext_vector_type(8))) float v8f;

#define NQ      16
#define NSTATES 65536      // 2^16
#define NLAYERS 3
#define BATCH   256
#define QTHREADS 1024

// ---------------------------------------------------------------------------
// GEMM1: H1 = relu(X(256x100) @ W1(100x64) + b1)   -- WMMA f32 16x16x4
// grid (16,4), block 32 (one wave per 16x16 tile), K = 25 chunks of 4
// A frag (16x4 f32): lanes 0-15 -> M=l16, {K=k0,K=k0+1}; lanes 16-31 -> {K=k0+2,k0+3}
// B frag (4x16 f32): lanes 0-15 -> N=l16, rows K=k0..; lanes 16-31 -> rows k0+2..
// D: VGPR r -> row (half*8 + r), col l16
// ---------------------------------------------------------------------------
__global__ __launch_bounds__(32) void qfm_gemm1(
    const float* __restrict__ X, const float* __restrict__ W1,
    const float* __restrict__ b1, float* __restrict__ H1)
{
    const int lane = threadIdx.x;
    const int mt   = blockIdx.x;       // 0..15
    const int nt   = blockIdx.y;       // 0..3
    const int half = lane >> 4;        // 0 or 1
    const int l16  = lane & 15;
    const int row  = mt * 16 + l16;
    const int col  = nt * 16 + l16;

    v8f acc = {};
    #pragma unroll 5
    for (int kk = 0; kk < 25; ++kk) {
        const int k0 = kk * 4 + half * 2;
        v2f a, b;
        a.x = X[row * 100 + k0];
        a.y = X[row * 100 + k0 + 1];
        b.x = W1[(k0    ) * 64 + col];
        b.y = W1[(k0 + 1) * 64 + col];
        acc = __builtin_amdgcn_wmma_f32_16x16x4_f32(
                  false, a, false, b, (short)0, acc, false, false);
    }
    const float bias = b1[col];
    #pragma unroll
    for (int r = 0; r < 8; ++r) {
        const int m = mt * 16 + half * 8 + r;
        H1[m * 64 + col] = fmaxf(acc[r] + bias, 0.0f);
    }
}

// ---------------------------------------------------------------------------
// GEMM2: H = tanh(H1(256x64) @ W2(64x16) + b2)  -- WMMA f32 16x16x4
// grid 16, block 32; single N tile; K = 16 chunks of 4
// ---------------------------------------------------------------------------
__global__ __launch_bounds__(32) void qfm_gemm2(
    const float* __restrict__ H1, const float* __restrict__ W2,
    const float* __restrict__ b2, float* __restrict__ H)
{
    const int lane = threadIdx.x;
    const int mt   = blockIdx.x;       // 0..15
    const int half = lane >> 4;
    const int l16  = lane & 15;
    const int row  = mt * 16 + l16;

    v8f acc = {};
    #pragma unroll
    for (int kk = 0; kk < 16; ++kk) {
        const int k0 = kk * 4 + half * 2;
        v2f a, b;
        a.x = H1[row * 64 + k0];
        a.y = H1[row * 64 + k0 + 1];
        b.x = W2[(k0    ) * 16 + l16];
        b.y = W2[(k0 + 1) * 16 + l16];
        acc = __builtin_amdgcn_wmma_f32_16x16x4_f32(
                  false, a, false, b, (short)0, acc, false, false);
    }
    const float bias = b2[l16];
    #pragma unroll
    for (int r = 0; r < 8; ++r) {
        const int m = mt * 16 + half * 8 + r;
        H[m * 16 + l16] = tanhf(acc[r] + bias);
    }
}

// ---------------------------------------------------------------------------
// Quantum circuit: one workgroup per batch element, state in 256KB LDS.
// rot: (3,16,3)  ent: (3,15)  h: (256,16)  out: (256,65536) complex64
// ---------------------------------------------------------------------------
__global__ __launch_bounds__(QTHREADS) void qfm_quantum(
    const float* __restrict__ h, const float* __restrict__ rot,
    const float* __restrict__ ent, float* __restrict__ out)
{
    extern __shared__ float amp[];          // NSTATES floats = 256KB
    const int b   = blockIdx.x;
    const int tid = threadIdx.x;

    // init |0...0>
    #pragma unroll
    for (int i = tid; i < NSTATES; i += QTHREADS) amp[i] = 0.0f;
    if (tid == 0) amp[0] = 1.0f;

    // per-batch angles, hoisted
    float hv[NQ];
    #pragma unroll
    for (int q = 0; q < NQ; ++q) hv[q] = h[b * NQ + q];

    __syncthreads();

    for (int layer = 0; layer < NLAYERS; ++layer) {
        // --- single-qubit rotations ---
        for (int q = 0; q < NQ; ++q) {
            const float* rp = rot + (layer * NQ + q) * 3;
            const float hq = hv[q];
            float sx, cx, sy, cy, sz, cz;
            sincosf(0.5f * rp[0] * hq, &sx, &cx);
            sincosf(0.5f * rp[1] * hq, &sy, &cy);
            sincosf(0.5f * rp[2] * hq, &sz, &cz);
            const float m00 =  cx * cy * cz;
            const float m01 = -sx * sy * sz;
            const float m10 =  sx * sy * cz;
            const float m11 =  cx * cy * sz;
            const int mask = 1 << q;

            #pragma unroll 4
            for (int t = tid; t < (NSTATES >> 1); t += QTHREADS) {
                const int low = t & (mask - 1);
                const int i0  = ((t >> q) << (q + 1)) | low;   // bit q == 0
                const int i1  = i0 | mask;
                const float a0 = amp[i0];
                const float a1 = amp[i1];
                amp[i0] = fmaf(m00, a0, m01 * a1);
                amp[i1] = fmaf(m10, a0, m11 * a1);
            }
            __syncthreads();
        }
        // --- noisy CNOTs: control q, target q+1 ---
        for (int q = 0; q < NQ - 1; ++q) {
            const float e = ent[layer * (NQ - 1) + q];
            const float p = 1.0f / (1.0f + __expf(-e));
            const float omp = 1.0f - p;
            const int cm = 1 << q;
            const int tm = 1 << (q + 1);

            // indices with control bit 1, target bit 0: 2^14 pairs
            #pragma unroll 4
            for (int t = tid; t < (NSTATES >> 2); t += QTHREADS) {
                const int low  = t & (cm - 1);
                const int rest = t >> q;
                const int i0   = (rest << (q + 2)) | cm | low;  // ctrl=1, tgt=0
                const int i1   = i0 | tm;
                const float a0 = amp[i0];
                const float a1 = amp[i1];
                amp[i0] = fmaf(omp, a0, p * a1);
                amp[i1] = fmaf(omp, a1, p * a0);
            }
            __syncthreads();
        }
    }

    // write complex64 interleaved (imag == 0), two states per b128 store
    float4* o = (float4*)out + (size_t)b * (NSTATES / 2);
    #pragma unroll
    for (int i = tid; i < NSTATES / 2; i += QTHREADS) {
        float4 v;
        v.x = amp[2 * i];     v.y = 0.0f;
        v.z = amp[2 * i + 1]; v.w = 0.0f;
        o[i] = v;
    }
}

// ---------------------------------------------------------------------------
extern "C" void kernel_launch(void* const* d_in, const int* in_sizes, int n_in,
                              void* d_out, int out_size, void* d_ws, size_t ws_size,
                              hipStream_t stream) {
    const float* X   = (const float*)d_in[0];  // molecular_features (256,100)
    const float* rot = (const float*)d_in[1];  // rotation_params    (3,16,3)
    const float* ent = (const float*)d_in[2];  // entangling_params  (3,15)
    const float* W1  = (const float*)d_in[3];  // (100,64)
    const float* b1  = (const float*)d_in[4];  // (64,)
    const float* W2  = (const float*)d_in[5];  // (64,16)
    const float* b2  = (const float*)d_in[6];  // (16,)
    float* out = (float*)d_out;                // (256,65536) complex64 -> float pairs

    float* H1 = (float*)d_ws;                  // 256*64 floats
    float* H  = H1 + BATCH * 64;               // 256*16 floats

    qfm_gemm1<<<dim3(16, 4), 32, 0, stream>>>(X, W1, b1, H1);
    qfm_gemm2<<<16, 32, 0, stream>>>(H1, W2, b2, H);
    qfm_quantum<<<BATCH, QTHREADS, NSTATES * sizeof(float), stream>>>(H, rot, ent, out);
}